// UpsamplerAttentionParallel_41918880809000
// MI455X (gfx1250) — compile-verified
//
#include <hip/hip_runtime.h>
#include <hip/hip_bf16.h>

typedef __attribute__((ext_vector_type(2))) float v2f;
typedef __attribute__((ext_vector_type(4))) float f4;
typedef __attribute__((ext_vector_type(8))) float v8f;

// D = A(16x4 f32) * B(4x16 f32) + C, fp32 WMMA
__device__ inline v8f wmma4(v2f a, v2f b, v8f c) {
  return __builtin_amdgcn_wmma_f32_16x16x4_f32(
      /*neg_a=*/false, a, /*neg_b=*/false, b,
      /*c_mod=*/(short)0, c, /*reuse_a=*/false, /*reuse_b=*/false);
}

// ---------------------------------------------------------------------------
// K1: Q = Wf*M + bf ; K = Wk*M + bk   (per batch: [64x128] x [128x4096])
// block = 256 thr = 8 waves; waves 0-3 -> row tiles, cols +0; waves 4-7 cols +16
// grid = (4096/32, B)
// ---------------------------------------------------------------------------
__global__ void qk_gemm(const float* __restrict__ M,
                        const float* __restrict__ Wf, const float* __restrict__ bf,
                        const float* __restrict__ Wk, const float* __restrict__ bk,
                        float* __restrict__ Q, float* __restrict__ K) {
  const int b    = blockIdx.y;
  const int lane = threadIdx.x & 31;
  const int wave = threadIdx.x >> 5;
  const int c0   = (wave & 3) * 16;
  const int p0   = blockIdx.x * 32 + (wave >> 2) * 16;
  const int mrow = lane & 15;          // A row (C) and B col (pixel)
  const int khi  = (lane >> 4) * 2;    // K sub-slot
  const float* Mb = M + (size_t)b * 128 * 4096;

  v8f accQ = {};
  v8f accK = {};
  for (int k = 0; k < 128; k += 4) {
    const int fk = k + khi;
    v2f af, ak, bb;
    af.x = Wf[(c0 + mrow) * 128 + fk];
    af.y = Wf[(c0 + mrow) * 128 + fk + 1];
    ak.x = Wk[(c0 + mrow) * 128 + fk];
    ak.y = Wk[(c0 + mrow) * 128 + fk + 1];
    bb.x = Mb[(size_t)fk * 4096 + p0 + mrow];
    bb.y = Mb[(size_t)(fk + 1) * 4096 + p0 + mrow];
    accQ = wmma4(af, bb, accQ);
    accK = wmma4(ak, bb, accK);
  }
  const int p = p0 + mrow;
  const int mofs = (lane >> 4) * 8;
#pragma unroll
  for (int j = 0; j < 8; ++j) {
    const int c = c0 + j + mofs;
    Q[((size_t)b * 64 + c) * 4096 + p] = accQ[j] + bf[c];
    K[((size_t)b * 64 + c) * 4096 + p] = accK[j] + bk[c];
  }
}

// ---------------------------------------------------------------------------
// K2: qp[c,p] = sum_d Wc[d,c] * Q[d,p]   ([64x64]^T x [64x4096])
// grid = (4096/32, B)
// ---------------------------------------------------------------------------
__global__ void qp_gemm(const float* __restrict__ Q, const float* __restrict__ Wc,
                        float* __restrict__ QP) {
  const int b    = blockIdx.y;
  const int lane = threadIdx.x & 31;
  const int wave = threadIdx.x >> 5;
  const int c0   = (wave & 3) * 16;
  const int p0   = blockIdx.x * 32 + (wave >> 2) * 16;
  const int mrow = lane & 15;
  const int khi  = (lane >> 4) * 2;

  v8f acc = {};
  for (int k = 0; k < 64; k += 4) {
    const int dk = k + khi;
    v2f a, bb;
    a.x  = Wc[dk * 64 + (c0 + mrow)];        // transposed access
    a.y  = Wc[(dk + 1) * 64 + (c0 + mrow)];
    bb.x = Q[((size_t)b * 64 + dk) * 4096 + p0 + mrow];
    bb.y = Q[((size_t)b * 64 + dk + 1) * 4096 + p0 + mrow];
    acc = wmma4(a, bb, acc);
  }
  const int p = p0 + mrow;
  const int mofs = (lane >> 4) * 8;
#pragma unroll
  for (int j = 0; j < 8; ++j) {
    const int c = c0 + j + mofs;
    QP[((size_t)b * 64 + c) * 4096 + p] = acc[j];   // bq term cancels in softmax
  }
}

// ---------------------------------------------------------------------------
// K3: kq[p] = sum_c K[c,p] * qp[c,p]   grid = (4096/256, B)
// ---------------------------------------------------------------------------
__global__ void kq_dot(const float* __restrict__ K, const float* __restrict__ QP,
                       float* __restrict__ KQ) {
  const int b = blockIdx.y;
  const int p = blockIdx.x * 256 + threadIdx.x;
  float s = 0.f;
#pragma unroll 8
  for (int c = 0; c < 64; ++c)
    s += K[((size_t)b * 64 + c) * 4096 + p] * QP[((size_t)b * 64 + c) * 4096 + p];
  KQ[(size_t)b * 4096 + p] = s;
}

// ---------------------------------------------------------------------------
// K4: V = conv3x3(M, Vw) + Vb as 9 tap-shifted GEMMs on WMMA.
// grid = (2, 64, 8): x-half, y-row, batch.  Borders: clamped addr * mask
// (keeps EXEC all-ones for the WMMAs).
// ---------------------------------------------------------------------------
__global__ void conv_gemm(const float* __restrict__ M, const float* __restrict__ Vw,
                          const float* __restrict__ Vb, float* __restrict__ V) {
  const int b    = blockIdx.z;
  const int y    = blockIdx.y;
  const int lane = threadIdx.x & 31;
  const int wave = threadIdx.x >> 5;
  const int c0   = (wave & 3) * 16;
  const int x0   = blockIdx.x * 32 + (wave >> 2) * 16;
  const int mrow = lane & 15;
  const int khi  = (lane >> 4) * 2;
  const float* Mb = M + (size_t)b * 128 * 4096;

  v8f acc = {};
  for (int tap = 0; tap < 9; ++tap) {
    const int ky = tap / 3 - 1, kx = tap % 3 - 1;
    const int ys = y + ky;
    const int xs = x0 + mrow + kx;
    const float msk = (ys >= 0 && ys < 64 && xs >= 0 && xs < 64) ? 1.f : 0.f;
    const int ysc = min(max(ys, 0), 63);
    const int xsc = min(max(xs, 0), 63);
    const int pix = ysc * 64 + xsc;
    for (int f = 0; f < 128; f += 4) {
      const int fk = f + khi;
      v2f a, bb;
      a.x  = Vw[((c0 + mrow) * 128 + fk) * 9 + tap];
      a.y  = Vw[((c0 + mrow) * 128 + fk + 1) * 9 + tap];
      bb.x = Mb[(size_t)fk * 4096 + pix] * msk;
      bb.y = Mb[(size_t)(fk + 1) * 4096 + pix] * msk;
      acc = wmma4(a, bb, acc);
    }
  }
  const int p = y * 64 + x0 + mrow;
  const int mofs = (lane >> 4) * 8;
#pragma unroll
  for (int j = 0; j < 8; ++j) {
    const int c = c0 + j + mofs;
    V[((size_t)b * 64 + c) * 4096 + p] = acc[j] + Vb[c];
  }
}

// ---------------------------------------------------------------------------
// K5: fused attention over contexts (bandwidth bound). Tile: one hi-res row y,
// 32 x-pixels. contexts read once from HBM with NT b128 loads, twice from LDS;
// out written NT b128.
// grid = (4, 128, 8), 256 threads.
// ---------------------------------------------------------------------------
__global__ void attn_fused(const float* __restrict__ ctx, const float* __restrict__ QP,
                           const float* __restrict__ V, const float* __restrict__ KQ,
                           float* __restrict__ out) {
  __shared__ float sctx[3 * 64 * 33];   // [n][c][x(32)+pad]  (pad 33: conflict-free)
  __shared__ float sqp[64 * 17];        // [c][plo(16)+pad]
  __shared__ float sv[64 * 17];
  __shared__ float skq[16];
  __shared__ float sattn[32 * 4];

  const int b  = blockIdx.z;
  const int y  = blockIdx.y;
  const int x0 = blockIdx.x * 32;
  const int t  = threadIdx.x;

  // stage contexts: 1536 float4 loads (NT, streamed once), scalar LDS stores
  for (int i4 = t; i4 < 3 * 64 * 8; i4 += 256) {
    const int n   = i4 >> 9;          // /512
    const int rem = i4 & 511;
    const int c   = rem >> 3;
    const int x   = (rem & 7) * 4;
    const f4 v = __builtin_nontemporal_load((const f4*)(
        ctx + ((((size_t)b * 3 + n) * 64 + c) * 128 + y) * 128 + x0 + x));
    float* s = &sctx[(n * 64 + c) * 33 + x];
    s[0] = v.x; s[1] = v.y; s[2] = v.z; s[3] = v.w;
  }
  const int plo = (y >> 1) * 64 + (x0 >> 1);
  {
    const int c  = t >> 2;
    const int pl = (t & 3) * 4;
    const f4 q = *(const f4*)(QP + ((size_t)b * 64 + c) * 4096 + plo + pl);
    const f4 w = *(const f4*)(V  + ((size_t)b * 64 + c) * 4096 + plo + pl);
    float* sq = &sqp[c * 17 + pl];
    float* sw = &sv [c * 17 + pl];
    sq[0] = q.x; sq[1] = q.y; sq[2] = q.z; sq[3] = q.w;
    sw[0] = w.x; sw[1] = w.y; sw[2] = w.z; sw[3] = w.w;
  }
  if (t < 16) skq[t] = KQ[(size_t)b * 4096 + plo + t];
  __syncthreads();

  // logits + softmax (32 pixels)
  if (t < 32) {
    const int p = t, ph = p >> 1;
    float l0 = 0.f, l1 = 0.f, l2 = 0.f;
#pragma unroll 8
    for (int c = 0; c < 64; ++c) {
      const float q = sqp[c * 17 + ph];
      l0 += sctx[(0 * 64 + c) * 33 + p] * q;
      l1 += sctx[(1 * 64 + c) * 33 + p] * q;
      l2 += sctx[(2 * 64 + c) * 33 + p] * q;
    }
    const float l3 = skq[ph];
    const float mx = fmaxf(fmaxf(l0, l1), fmaxf(l2, l3));
    const float e0 = __expf(l0 - mx), e1 = __expf(l1 - mx);
    const float e2 = __expf(l2 - mx), e3 = __expf(l3 - mx);
    const float inv = 1.f / (e0 + e1 + e2 + e3);
    sattn[p * 4 + 0] = e0 * inv;
    sattn[p * 4 + 1] = e1 * inv;
    sattn[p * 4 + 2] = e2 * inv;
    sattn[p * 4 + 3] = e3 * inv;
  }
  __syncthreads();

  // output: thread t -> channel c = t>>2, x-chunk of 8; NT float4 stores
  const int c  = t >> 2;
  const int xq = (t & 3) * 8;
  float* ob = out + (((size_t)b * 64 + c) * 128 + y) * 128 + x0;
#pragma unroll
  for (int j4 = 0; j4 < 2; ++j4) {
    const int xb = xq + j4 * 4;
    f4 r;
#pragma unroll
    for (int jj = 0; jj < 4; ++jj) {
      const int x = xb + jj;
      const float a0 = sattn[x * 4 + 0], a1 = sattn[x * 4 + 1];
      const float a2 = sattn[x * 4 + 2], a3 = sattn[x * 4 + 3];
      r[jj] = a0 * sctx[(0 * 64 + c) * 33 + x] +
              a1 * sctx[(1 * 64 + c) * 33 + x] +
              a2 * sctx[(2 * 64 + c) * 33 + x] +
              a3 * sv[c * 17 + (x >> 1)];
    }
    __builtin_nontemporal_store(r, (f4*)(ob + xb));
  }
}

extern "C" void kernel_launch(void* const* d_in, const int* in_sizes, int n_in,
                              void* d_out, int out_size, void* d_ws, size_t ws_size,
                              hipStream_t stream) {
  (void)in_sizes; (void)n_in; (void)out_size; (void)ws_size;
  const float* ctx = (const float*)d_in[0];   // [8,3,64,128,128]
  const float* M   = (const float*)d_in[1];   // [8,128,64,64]
  const float* Wc  = (const float*)d_in[2];   // [64,64]
  const float* bc  = (const float*)d_in[3];   // cancels in softmax
  const float* Wf  = (const float*)d_in[4];   // [64,128]
  const float* bf  = (const float*)d_in[5];
  const float* Wk  = (const float*)d_in[6];   // [64,128]
  const float* bk  = (const float*)d_in[7];
  const float* Vw  = (const float*)d_in[8];   // [64,128,3,3]
  const float* Vb  = (const float*)d_in[9];
  (void)bc;
  float* out = (float*)d_out;                 // [8,64,128,128]

  float* ws = (float*)d_ws;
  float* Q  = ws;                 // 8*64*4096
  float* K  = ws + 2097152;
  float* QP = ws + 4194304;
  float* V  = ws + 6291456;
  float* KQ = ws + 8388608;       // 8*4096

  qk_gemm   <<<dim3(128, 8),     256, 0, stream>>>(M, Wf, bf, Wk, bk, Q, K);
  qp_gemm   <<<dim3(128, 8),     256, 0, stream>>>(Q, Wc, QP);
  kq_dot    <<<dim3(16, 8),      256, 0, stream>>>(K, QP, KQ);
  conv_gemm <<<dim3(2, 64, 8),   256, 0, stream>>>(M, Vw, Vb, V);
  attn_fused<<<dim3(4, 128, 8),  256, 0, stream>>>(ctx, QP, V, KQ, out);
}